// OkoSetLossHardK_90254442758162
// MI455X (gfx1250) — compile-verified
//
#include <hip/hip_runtime.h>
#include <hip/hip_bf16.h>

#define B_N 8192
#define C_N 1024

typedef __attribute__((ext_vector_type(2))) float v2f;
typedef __attribute__((ext_vector_type(8))) float v8f;

// ---------------- phase 0: first/second occurrence per class ----------------
__global__ void k_init(int* __restrict__ first, int* __restrict__ second) {
    int c = blockIdx.x * 256 + threadIdx.x;
    if (c < C_N) { first[c] = 0x7FFFFFFF; second[c] = 0x7FFFFFFF; }
}

__global__ void k_occ1(const int* __restrict__ target, int* __restrict__ first) {
    int i = blockIdx.x * 256 + threadIdx.x;
    if (i < B_N) atomicMin(&first[target[i]], i);
}

__global__ void k_occ2(const int* __restrict__ target, const int* __restrict__ first,
                       int* __restrict__ second) {
    int i = blockIdx.x * 256 + threadIdx.x;
    if (i < B_N) {
        int t = target[i];
        if (first[t] != i) atomicMin(&second[t], i);
    }
}

// ---------------- phase 1: L[j] = logsumexp(x[j,:]) --------------------------
// one wave per row; row cached in 8 float4 registers (1024 floats / 32 lanes)
__global__ void k_rowlse(const float* __restrict__ x, float* __restrict__ L) {
    int lane = threadIdx.x & 31;
    int wv   = threadIdx.x >> 5;
    int j    = blockIdx.x * 8 + wv;
    const float4* row = (const float4*)(x + (size_t)j * C_N);
    float4 r[8];
#pragma unroll
    for (int k = 0; k < 8; ++k) r[k] = row[lane + 32 * k];   // global_load_b128, coalesced
    float m = -INFINITY;
#pragma unroll
    for (int k = 0; k < 8; ++k)
        m = fmaxf(m, fmaxf(fmaxf(r[k].x, r[k].y), fmaxf(r[k].z, r[k].w)));
#pragma unroll
    for (int off = 16; off >= 1; off >>= 1) m = fmaxf(m, __shfl_xor(m, off, 32));
    float s = 0.f;
#pragma unroll
    for (int k = 0; k < 8; ++k)
        s += __expf(r[k].x - m) + __expf(r[k].y - m) + __expf(r[k].z - m) + __expf(r[k].w - m);
#pragma unroll
    for (int off = 16; off >= 1; off >>= 1) s += __shfl_xor(s, off, 32);
    if (lane == 0) L[j] = m + __logf(s);
}

// ---------------- phase 2: per-class hardest negative ------------------------
// negc[c] = argmin_{j: target[j]!=c} (x[j,c] - L[j]); partials over 16 row-chunks
__global__ void k_negpart(const float* __restrict__ x, const float* __restrict__ L,
                          const int* __restrict__ target,
                          float* __restrict__ pvals, int* __restrict__ pidx) {
    __shared__ float sL[512];
    __shared__ int   sT[512];
    __shared__ float rv[8][32];
    __shared__ int   ri[8][32];
    const int tx = threadIdx.x, ty = threadIdx.y;
    const int tid = ty * 32 + tx;
    const int j0 = blockIdx.y * 512;
    for (int r = tid; r < 512; r += 256) { sL[r] = L[j0 + r]; sT[r] = target[j0 + r]; }
    __syncthreads();
    const int c = blockIdx.x * 32 + tx;
    float best = INFINITY; int bestj = 0x7FFFFFFF;
    for (int jj = ty; jj < 512; jj += 8) {
        int j = j0 + jj;
        if (jj + 64 < 512)
            __builtin_prefetch(&x[(size_t)(j + 64) * C_N + c], 0, 1);   // global_prefetch_b8
        float s = x[(size_t)j * C_N + c] - sL[jj];
        if (sT[jj] != c && s < best) { best = s; bestj = j; }
    }
    rv[ty][tx] = best; ri[ty][tx] = bestj;
    __syncthreads();
    if (ty == 0) {
#pragma unroll
        for (int q = 1; q < 8; ++q) {
            float v = rv[q][tx]; int id = ri[q][tx];
            if (v < best || (v == best && id < bestj)) { best = v; bestj = id; }
        }
        pvals[blockIdx.y * C_N + c] = best;
        pidx [blockIdx.y * C_N + c] = bestj;
    }
}

__global__ void k_negfin(const float* __restrict__ pvals, const int* __restrict__ pidx,
                         int* __restrict__ negc) {
    int c = blockIdx.x * 256 + threadIdx.x;
    float best = INFINITY; int bestj = 0x7FFFFFFF;
#pragma unroll
    for (int q = 0; q < 16; ++q) {
        float v = pvals[q * C_N + c]; int id = pidx[q * C_N + c];
        if (v < best || (v == best && id < bestj)) { best = v; bestj = id; }
    }
    negc[c] = bestj;
}

// ---------------- phase 3: per-anchor cross-entropy of triplet sum -----------
// One wave = 16 anchors. Σexp accumulated by v_wmma_f32_16x16x4_f32 (A = exp
// values in the ISA 16x4 layout, B = ones) so the adds run on the XDL pipe and
// co-execute with the v_exp_f32 TRANS stream.
__global__ void k_phase3(const float* __restrict__ x, const int* __restrict__ target,
                         const int* __restrict__ first, const int* __restrict__ second,
                         const int* __restrict__ negc, float* __restrict__ loss_i) {
    const int lane = threadIdx.x & 31;
    const int wv   = threadIdx.x >> 5;
    const int half = lane >> 4;                 // 0: K=0,1   1: K=2,3
    const int a    = blockIdx.x * 128 + wv * 16 + (lane & 15);  // anchor (row M)
    const int t  = target[a];
    const int fi = first[t];
    const int p  = (fi == a) ? second[t] : fi;
    const int n  = negc[t];
    const float* xi = x + (size_t)a * C_N;
    const float* xp = x + (size_t)p * C_N;
    const float* xn = x + (size_t)n * C_N;

    // sweep 1: row max of v = xi + xp + xn
    float m = -INFINITY;
#pragma unroll 4
    for (int k = 0; k < 256; ++k) {
        int c0 = (k << 2) + (half << 1);
        v2f vi = *(const v2f*)(xi + c0);
        v2f vp = *(const v2f*)(xp + c0);
        v2f vn = *(const v2f*)(xn + c0);
        m = fmaxf(m, fmaxf(vi[0] + vp[0] + vn[0], vi[1] + vp[1] + vn[1]));
    }
    m = fmaxf(m, __shfl_xor(m, 16, 32));        // combine K-halves of same anchor

    // sweep 2: sum of exp via WMMA accumulation (uniform control flow, EXEC all 1s)
    v8f acc = {};
    v2f bones; bones[0] = 1.0f; bones[1] = 1.0f;
#pragma unroll 4
    for (int k = 0; k < 256; ++k) {
        int c0 = (k << 2) + (half << 1);
        v2f vi = *(const v2f*)(xi + c0);
        v2f vp = *(const v2f*)(xp + c0);
        v2f vn = *(const v2f*)(xn + c0);
        v2f ae;
        ae[0] = __expf(vi[0] + vp[0] + vn[0] - m);
        ae[1] = __expf(vi[1] + vp[1] + vn[1] - m);
        acc = __builtin_amdgcn_wmma_f32_16x16x4_f32(
            /*neg_a=*/false, ae, /*neg_b=*/false, bones,
            /*c_mod=*/(short)0, acc, /*reuse_a=*/false, /*reuse_b=*/false);
    }
    // D layout: lanes 0-15 VGPR g = row M=g; lanes 16-31 VGPR g = row M=g+8.
    float tmp   = acc[lane & 7];
    float other = __shfl_xor(tmp, 16, 32);
    float sumexp = (lane & 8) ? other : tmp;

    float vt = xi[t] + xp[t] + xn[t];
    if (half == 0) loss_i[a] = (m + __logf(sumexp)) - vt;   // = -logp[a, t]
}

// ---------------- final deterministic mean -----------------------------------
__global__ void k_reduce(const float* __restrict__ loss_i, float* __restrict__ out) {
    __shared__ float sm[256];
    float s = 0.f;
    for (int i = threadIdx.x; i < B_N; i += 256) s += loss_i[i];
    sm[threadIdx.x] = s;
    __syncthreads();
    for (int off = 128; off >= 1; off >>= 1) {
        if (threadIdx.x < off) sm[threadIdx.x] += sm[threadIdx.x + off];
        __syncthreads();
    }
    if (threadIdx.x == 0) out[0] = sm[0] / (float)B_N;
}

extern "C" void kernel_launch(void* const* d_in, const int* in_sizes, int n_in,
                              void* d_out, int out_size, void* d_ws, size_t ws_size,
                              hipStream_t stream) {
    const float* x      = (const float*)d_in[0];
    const int*   target = (const int*)d_in[1];

    // workspace carve-out (~208 KB)
    float* wsf    = (float*)d_ws;
    float* L      = wsf;                          // B floats
    int*   first  = (int*)(wsf + B_N);            // C ints
    int*   second = first + C_N;                  // C ints
    int*   negc   = second + C_N;                 // C ints
    float* pvals  = (float*)(negc + C_N);         // 16*C floats
    int*   pidx   = (int*)(pvals + 16 * C_N);     // 16*C ints
    float* loss_i = (float*)(pidx + 16 * C_N);    // B floats

    k_init   <<<dim3(4),        dim3(256),    0, stream>>>(first, second);
    k_occ1   <<<dim3(32),       dim3(256),    0, stream>>>(target, first);
    k_occ2   <<<dim3(32),       dim3(256),    0, stream>>>(target, first, second);
    k_rowlse <<<dim3(1024),     dim3(256),    0, stream>>>(x, L);
    k_negpart<<<dim3(32, 16),   dim3(32, 8),  0, stream>>>(x, L, target, pvals, pidx);
    k_negfin <<<dim3(4),        dim3(256),    0, stream>>>(pvals, pidx, negc);
    k_phase3 <<<dim3(64),       dim3(256),    0, stream>>>(x, target, first, second, negc, loss_i);
    k_reduce <<<dim3(1),        dim3(256),    0, stream>>>(loss_i, (float*)d_out);
}